// PixelVectorExtractor_25297357373688
// MI455X (gfx1250) — compile-verified
//
#include <hip/hip_runtime.h>
#include <stdint.h>

// ---------------------------------------------------------------------------
// PixelVectorExtractor for MI455X (gfx1250).
// Pure data-expansion op: 144 KB in -> 535.8 MB out  =>  store-bandwidth bound
// (~23 us floor at 23.3 TB/s). Strategy: stage x[n] in LDS via CDNA5 async
// global->LDS DMA (global_load_async_to_lds_b128 + s_wait_asynccnt), gather
// from LDS, and emit aligned non-temporal b128 stores over row *pairs*
// (pair base is always 16B-aligned even though odd rows alone are not).
// ---------------------------------------------------------------------------

#define AS1 __attribute__((address_space(1)))
#define AS3 __attribute__((address_space(3)))

#if defined(__has_builtin)
#if __has_builtin(__builtin_amdgcn_global_load_async_to_lds_b128)
#define HAVE_ASYNC_LDS 1
#endif
#if __has_builtin(__builtin_amdgcn_s_wait_asynccnt)
#define HAVE_WAIT_ASYNC 1
#endif
#endif

typedef __attribute__((ext_vector_type(4))) float v4f;
typedef int v4i __attribute__((vector_size(16)));   // matches async builtin's pointee type

namespace {
constexpr int kC        = 10;            // output channels (11th synthesized channel is sliced off)
constexpr int kHW       = 30;            // H == W
constexpr int kImg      = kHW * kHW;     // 900 floats per channel
constexpr int kChanImg  = kC * kImg;     // 9000 floats per image n (36 KB)
constexpr int kPos      = 61 * 61;       // 3721 canvas positions
constexpr int kRowElems = kC * kPos;     // 37210 floats per output row
constexpr int kPairElems = 2 * kRowElems;            // 74420 = 4 * 18605 -> exact b128 tiling
constexpr int kRows     = 4 * kHW * kHW; // 3600 output rows
constexpr int kThreads  = 256;           // 8 waves (wave32)
}

// Value of output element at flat offset f (= c*3721 + i*61 + j) of row (h,w).
__device__ __forceinline__ float pve_elem(const float* lx, unsigned f, int h, int w) {
    unsigned c   = f / 3721u;            // channel 0..9
    unsigned pos = f - c * 3721u;
    unsigned i   = pos / 61u;            // 0..60
    unsigned j   = pos - i * 61u;        // 0..60
    if (i >= 59u || j >= 59u) return 0.0f;               // pad frame to 61x61
    int ih = h + (int)i - 29;                            // source pixel
    int iw = w + (int)j - 29;
    if ((unsigned)ih < 30u && (unsigned)iw < 30u)
        return lx[c * 900u + (unsigned)ih * 30u + (unsigned)iw];
    return (c == 0u) ? 1.0f : 0.0f;                      // border constant
}

__global__ __launch_bounds__(kThreads) void PixelVectorExtractor_25297357373688_kernel(
        const float* __restrict__ x, float* __restrict__ out) {
    __shared__ float lx[kChanImg];                       // 36 KB: whole image n, all channels

    const int row0 = blockIdx.x * 2;                     // even row -> pair base 16B-aligned
    const int row1 = row0 + 1;
    const int n    = row0 / 900;                         // 900 even -> both rows share n
    const float* xn = x + n * kChanImg;

#if defined(HAVE_ASYNC_LDS)
    // CDNA5 async DMA path: GLOBAL_LOAD_ASYNC_TO_LDS_B128, tracked by ASYNCcnt.
    for (int k = threadIdx.x; k < kChanImg / 4; k += kThreads) {
        AS1 v4i* g = (AS1 v4i*)(unsigned long long)(const void*)(xn + 4 * k);
        // Generic LDS addresses carry the LDS byte offset in their low 32 bits.
        AS3 v4i* l = (AS3 v4i*)(unsigned)(unsigned long long)(const void*)&lx[4 * k];
        __builtin_amdgcn_global_load_async_to_lds_b128(g, l, 0, 0);
    }
#if defined(HAVE_WAIT_ASYNC)
    __builtin_amdgcn_s_wait_asynccnt(0);                 // this wave's async writes landed
#endif
#else
    // Fallback: cooperative b128 copy through VGPRs (ds_store_b128).
    {
        const v4f* src4 = (const v4f*)xn;                // 16B aligned (9000*4 % 16 == 0)
        v4f* dst4 = (v4f*)lx;
        for (int k = threadIdx.x; k < kChanImg / 4; k += kThreads)
            dst4[k] = src4[k];
    }
#endif
    __syncthreads();                                     // all waves' LDS fills visible

    const int h0 = (row0 / 30) % 30, w0 = row0 % 30;
    const int h1 = (row1 / 30) % 30, w1 = row1 % 30;
    float* opair = out + (size_t)row0 * kRowElems;       // byte offset row0*148840 ≡ 0 (mod 16)

    // Flat-index the two-row region: 74420 floats = 18605 aligned float4 stores.
    // Consecutive lanes -> consecutive 16B chunks -> 512B contiguous per wave store.
    for (int q = threadIdx.x; q < kPairElems / 4; q += kThreads) {
        unsigned fbase = 4u * (unsigned)q;
        v4f v;
#pragma unroll
        for (int e = 0; e < 4; ++e) {
            unsigned f = fbase + (unsigned)e;
            bool second = f >= (unsigned)kRowElems;      // which row of the pair
            unsigned fl = second ? f - (unsigned)kRowElems : f;
            v[e] = pve_elem(lx, fl, second ? h1 : h0, second ? w1 : w0);
        }
        // Output (536 MB) far exceeds 192 MB L2 -> non-temporal b128 stores.
        __builtin_nontemporal_store(v, (v4f*)(opair + fbase));
    }
}

extern "C" void kernel_launch(void* const* d_in, const int* in_sizes, int n_in,
                              void* d_out, int out_size, void* d_ws, size_t ws_size,
                              hipStream_t stream) {
    (void)in_sizes; (void)n_in; (void)out_size; (void)d_ws; (void)ws_size;
    const float* x = (const float*)d_in[0];   // (4,10,30,30) f32, 36000 elements
    float* out = (float*)d_out;               // (3600,10,3721) f32, 133,956,000 elements
    dim3 grid(kRows / 2);                     // 1800 blocks, one row-pair each
    PixelVectorExtractor_25297357373688_kernel<<<grid, kThreads, 0, stream>>>(x, out);
}